// ReEig_22643067584717
// MI455X (gfx1250) — compile-verified
//
#include <hip/hip_runtime.h>

// ReEig on [16384, 64, 64] SPD fp32 matrices.
// out = eps*I + (M + M*sign(M))/2,  M = X - eps*I   (== U max(L,eps) U^T)
// sign(M) via scaled Newton-Schulz polynomial iterations -> pure 64x64 GEMMs
// -> v_wmma_f32_16x16x4_f32, one matrix per block, 16 wave32 waves per block.
//
// All operands in this algorithm are symmetric (polynomials of M), so the
// B operand is read via B[k][n] == B[n][k]: row-contiguous float2 loads,
// identical addressing pattern to A -> mergeable ds_load_2addr_b64 on both
// operands (halves B-side LDS instruction count vs column-strided b32 pairs).

#define LDSS 68           // padded LDS row stride (floats): kills bank conflicts
#define THRESH 1e-4f
#define QUINTIC_ITERS 8
#define CUBIC_ITERS 6

typedef __attribute__((ext_vector_type(2))) float v2f;
typedef __attribute__((ext_vector_type(8))) float v8f;

// C = A(64x64) * B(64x64), B symmetric; this wave's 16x16 tile (tm, tn).
// WMMA f32 16x16x4 operand layout (ISA 7.12.2, 32-bit):
//   A: lane g*16+m holds A[m][k+2g], A[m][k+2g+1]
//   B: VGPR r, lane g*16+n holds B[k+r+2g][n]  (== B[n][k+r+2g] by symmetry)
__device__ __forceinline__ v8f wmma_gemm64_sym(const float* __restrict__ A,
                                               const float* __restrict__ B,
                                               int tm, int tn, int lid, int half) {
  v8f acc = {};
  const float* arow = A + (tm * 16 + lid) * LDSS + 2 * half;
  const float* brow = B + (tn * 16 + lid) * LDSS + 2 * half;  // symmetric fetch
#pragma unroll
  for (int k = 0; k < 64; k += 4) {
    v2f a, b;
    a[0] = arow[k];
    a[1] = arow[k + 1];
    b[0] = brow[k];
    b[1] = brow[k + 1];
    acc = __builtin_amdgcn_wmma_f32_16x16x4_f32(
        /*neg_a=*/false, a, /*neg_b=*/false, b,
        /*c_mod=*/(short)0, acc, /*reuse_a=*/false, /*reuse_b=*/false);
  }
  return acc;
}

// C/D layout: VGPR v, lane g*16+n -> element (row = v + 8g, col = n)
__device__ __forceinline__ void store_tile(float* __restrict__ D, v8f acc,
                                           int tm, int tn, int lid, int half) {
  float* dp = D + (tm * 16 + 8 * half) * LDSS + tn * 16 + lid;
#pragma unroll
  for (int v = 0; v < 8; ++v) dp[v * LDSS] = acc[v];
}

// Load M = X - eps*I into padded LDS buffer; return this thread's sum of squares.
__device__ __forceinline__ float load_M(const float* __restrict__ Xb,
                                        float* __restrict__ dst, int tid) {
  float ss = 0.f;
#pragma unroll
  for (int p = 0; p < 2; ++p) {
    int flat = p * 2048 + tid * 4;
    int r = flat >> 6, c = flat & 63;
    float4 x = *(const float4*)(Xb + flat);
    float e0 = x.x - ((c + 0) == r ? THRESH : 0.f);
    float e1 = x.y - ((c + 1) == r ? THRESH : 0.f);
    float e2 = x.z - ((c + 2) == r ? THRESH : 0.f);
    float e3 = x.w - ((c + 3) == r ? THRESH : 0.f);
    float* q = dst + r * LDSS + c;
    q[0] = e0; q[1] = e1; q[2] = e2; q[3] = e3;
    ss += e0 * e0 + e1 * e1 + e2 * e2 + e3 * e3;
  }
  return ss;
}

__global__ __launch_bounds__(512) void reeig_sign_kernel(
    const float* __restrict__ X, float* __restrict__ out) {
  __shared__ float bufA[64 * LDSS];
  __shared__ float bufB[64 * LDSS];
  __shared__ float bufC[64 * LDSS];
  __shared__ float red[17];

  const int tid  = threadIdx.x;
  const int wave = tid >> 5;        // 16 waves
  const int lane = tid & 31;
  const int lid  = lane & 15;
  const int half = lane >> 4;
  const int tm   = wave >> 2;       // tile row 0..3
  const int tn   = wave & 3;        // tile col 0..3

  const float* Xb = X + (size_t)blockIdx.x * 4096;
  float* Ob = out + (size_t)blockIdx.x * 4096;

  // ---- Y0 = (X - eps I) / ||X - eps I||_F  (spectral norm <= Frobenius) ----
  float ss = load_M(Xb, bufA, tid);
#pragma unroll
  for (int off = 16; off; off >>= 1) ss += __shfl_xor(ss, off, 32);
  if (lane == 0) red[wave] = ss;
  __syncthreads();
  if (tid == 0) {                    // deterministic serial combine (no atomics)
    float t = 0.f;
#pragma unroll
    for (int w = 0; w < 16; ++w) t += red[w];
    red[16] = rsqrtf(t);
  }
  __syncthreads();
  const float s = red[16];
#pragma unroll
  for (int p = 0; p < 2; ++p) {      // scale in place
    int flat = p * 2048 + tid * 4;
    float* q = &bufA[(flat >> 6) * LDSS + (flat & 63)];
    q[0] *= s; q[1] *= s; q[2] *= s; q[3] *= s;
  }

  float* Y = bufA;  // current iterate (-> sign(M))
  float* Z = bufB;  // Y^2 / next iterate
  float* T = bufC;  // polynomial P

  const int orow = tm * 16 + 8 * half;   // this lane's first output row
  const int ocol = tn * 16 + lid;

  // ---- aggressive quintic phase: Y <- Y*(a I + b Y^2 + c Y^4) ----
  const float qa = 3.4445f, qb = -4.7750f, qc = 2.0315f;
  for (int it = 0; it < QUINTIC_ITERS; ++it) {
    __syncthreads();
    v8f z = wmma_gemm64_sym(Y, Y, tm, tn, lid, half);   // Z = Y^2
    store_tile(Z, z, tm, tn, lid, half);
    __syncthreads();
    v8f t4 = wmma_gemm64_sym(Z, Z, tm, tn, lid, half);  // Z^2 (= Y^4)
    {                                                    // T = qa I + qb Z + qc Z^2
      float* dp = T + orow * LDSS + ocol;
      const float* zp = Z + orow * LDSS + ocol;
#pragma unroll
      for (int v = 0; v < 8; ++v) {
        float val = qb * zp[v * LDSS] + qc * t4[v];
        if ((orow + v) == ocol) val += qa;
        dp[v * LDSS] = val;
      }
    }
    __syncthreads();
    v8f yn = wmma_gemm64_sym(Y, T, tm, tn, lid, half);  // Y' = Y * T
    store_tile(Z, yn, tm, tn, lid, half);
    float* tmp = Y; Y = Z; Z = tmp;
  }

  // ---- cubic Newton-Schulz polish: Y <- Y*(1.5 I - 0.5 Y^2) ----
  for (int it = 0; it < CUBIC_ITERS; ++it) {
    __syncthreads();
    v8f z = wmma_gemm64_sym(Y, Y, tm, tn, lid, half);   // Y^2, fused into P
    {
      float* dp = T + orow * LDSS + ocol;
#pragma unroll
      for (int v = 0; v < 8; ++v) {
        float val = -0.5f * z[v];
        if ((orow + v) == ocol) val += 1.5f;
        dp[v * LDSS] = val;
      }
    }
    __syncthreads();
    v8f yn = wmma_gemm64_sym(Y, T, tm, tn, lid, half);
    store_tile(Z, yn, tm, tn, lid, half);
    float* tmp = Y; Y = Z; Z = tmp;
  }

  // ---- out = eps I + 0.5*(M + M*S), S = Y ----
  __syncthreads();
  (void)load_M(Xb, T, tid);          // reload M (cheaper than a 4th LDS buffer)
  __syncthreads();
  v8f g = wmma_gemm64_sym(T, Y, tm, tn, lid, half);     // G = M * S
  {
    const float* mp = T + orow * LDSS + ocol;
    float* op = Ob + orow * 64 + ocol;
#pragma unroll
    for (int v = 0; v < 8; ++v) {
      float val = 0.5f * (mp[v * LDSS] + g[v]);
      if ((orow + v) == ocol) val += THRESH;
      op[v * 64] = val;
    }
  }
}

extern "C" void kernel_launch(void* const* d_in, const int* in_sizes, int n_in,
                              void* d_out, int out_size, void* d_ws, size_t ws_size,
                              hipStream_t stream) {
  const float* X = (const float*)d_in[0];
  float* O = (float*)d_out;
  const int batch = in_sizes[0] / 4096;  // 16384 matrices of 64x64
  reeig_sign_kernel<<<dim3(batch), dim3(512), 0, stream>>>(X, O);
}